// E2Net_36618891165869
// MI455X (gfx1250) — compile-verified
//
#include <hip/hip_runtime.h>
#include <hip/hip_bf16.h>

// ---------------------------------------------------------------------------
// Spiking net (LIF, subtract reset) on MI455X / gfx1250.
//   per step:  cur = x_t @ w1m^T ; LIF1 ; out = spk1 @ w2m^T ; LIF2 -> spikes
// Both GEMMs run on V_WMMA_F32_16X16X4_F32 (full fp32 — Heaviside thresholds
// make low-precision membrane potentials risky). NT layout => A and B frags
// load with identical single b64-per-lane pattern, no transpose / LDS needed;
// weights + state (26 MB) live in the 192 MB L2 across all 20 steps.
// ---------------------------------------------------------------------------

typedef __attribute__((ext_vector_type(2))) float v2f;
typedef __attribute__((ext_vector_type(8))) float v8f;

#define TT 20
#define BB 1024
#define FF 512
#define HH 2048
#define CC 512

#define THRESH 1.0f
#define BETA   0.9f

// ---- init: masked weights + zero membranes (must re-run every call) -------
__global__ void e2net_init(const float* __restrict__ w1, const float* __restrict__ m1,
                           const float* __restrict__ w2, const float* __restrict__ m2,
                           float* __restrict__ w1m, float* __restrict__ w2m,
                           float* __restrict__ mem1, float* __restrict__ mem2) {
    unsigned i = blockIdx.x * blockDim.x + threadIdx.x;   // grid = B*H = 2097152
    if (i < (unsigned)(HH * FF)) {                        // 1048576: both weight mats
        w1m[i] = w1[i] * m1[i];
        w2m[i] = w2[i] * m2[i];
    }
    mem1[i] = 0.0f;                                       // B*H == grid size
    if (i < (unsigned)(BB * CC)) mem2[i] = 0.0f;          // 524288
}

// ---- fused LIF epilogue on one 16x16 f32 accumulator fragment -------------
// C/D layout: lanes 0-15 -> M = vgpr, N = lane ; lanes 16-31 -> M = 8+vgpr.
__device__ __forceinline__ void lif_store(v8f acc, float* __restrict__ mem,
                                          float* __restrict__ spk,
                                          int m0, int n0, int ldn, int lane) {
    const int n     = n0 + (lane & 15);
    const int mBase = m0 + ((lane >> 4) << 3);
#pragma unroll
    for (int i = 0; i < 8; ++i) {
        const size_t idx = (size_t)(mBase + i) * (size_t)ldn + (size_t)n;
        const float m_old = mem[idx];
        const float reset = (m_old > THRESH) ? 1.0f : 0.0f;       // detached spike of prev mem
        const float m_new = BETA * m_old + acc[i] - reset * THRESH;
        mem[idx] = m_new;
        spk[idx] = (m_new > THRESH) ? 1.0f : 0.0f;                // Heaviside(mem - thresh)
    }
}

// ---- GEMM (cur = A(MxK) · W(NxK)^T) fused with LIF state update -----------
// One wave32 computes a 32x32 output tile: 2x2 WMMA fragments, K-step of 4.
__global__ void lif_gemm(const float* __restrict__ A, const float* __restrict__ W,
                         float* __restrict__ mem, float* __restrict__ spk,
                         int M, int N, int K) {
    const int lane = threadIdx.x & 31;
    const int wave = threadIdx.x >> 5;
    const int tile = blockIdx.x * (blockDim.x >> 5) + wave;   // wave-uniform
    const int tilesN = N >> 5;
    const int tilesTotal = (M >> 5) * tilesN;
    if (tile >= tilesTotal) return;                           // whole-wave exit: EXEC stays full

    const int m0 = (tile / tilesN) << 5;
    const int n0 = (tile % tilesN) << 5;

    v8f acc00 = {}, acc01 = {}, acc10 = {}, acc11 = {};

    // A-frag 16x4 and B-frag 4x16 share the same per-lane addressing in NT form:
    // lane L reads row (base + (L&15)), K-pair k + 2*(L>>4) as one b64 load.
    const int   rlo = lane & 15;
    const int   khi = (lane >> 4) << 1;
    const float* a0p = A + (size_t)(m0 + rlo)      * (size_t)K + khi;
    const float* a1p = A + (size_t)(m0 + 16 + rlo) * (size_t)K + khi;
    const float* b0p = W + (size_t)(n0 + rlo)      * (size_t)K + khi;
    const float* b1p = W + (size_t)(n0 + 16 + rlo) * (size_t)K + khi;

#pragma unroll 4
    for (int k = 0; k < K; k += 4) {
        const v2f a0 = *(const v2f*)(a0p + k);
        const v2f a1 = *(const v2f*)(a1p + k);
        const v2f b0 = *(const v2f*)(b0p + k);
        const v2f b1 = *(const v2f*)(b1p + k);
        // 8 args: (neg_a, A, neg_b, B, c_mod, C, reuse_a, reuse_b)
        acc00 = __builtin_amdgcn_wmma_f32_16x16x4_f32(false, a0, false, b0, (short)0, acc00, false, false);
        acc01 = __builtin_amdgcn_wmma_f32_16x16x4_f32(false, a0, false, b1, (short)0, acc01, false, false);
        acc10 = __builtin_amdgcn_wmma_f32_16x16x4_f32(false, a1, false, b0, (short)0, acc10, false, false);
        acc11 = __builtin_amdgcn_wmma_f32_16x16x4_f32(false, a1, false, b1, (short)0, acc11, false, false);
    }

    lif_store(acc00, mem, spk, m0,      n0,      N, lane);
    lif_store(acc01, mem, spk, m0,      n0 + 16, N, lane);
    lif_store(acc10, mem, spk, m0 + 16, n0,      N, lane);
    lif_store(acc11, mem, spk, m0 + 16, n0 + 16, N, lane);
}

extern "C" void kernel_launch(void* const* d_in, const int* in_sizes, int n_in,
                              void* d_out, int out_size, void* d_ws, size_t ws_size,
                              hipStream_t stream) {
    const float* x     = (const float*)d_in[0];   // (T,B,F)
    const float* w1    = (const float*)d_in[1];   // (H,F)
    const float* w2    = (const float*)d_in[2];   // (C,H)
    const float* mask1 = (const float*)d_in[3];   // (H,F)
    const float* mask2 = (const float*)d_in[4];   // (C,H)
    float* out = (float*)d_out;                   // (T,B,C)
    float* ws  = (float*)d_ws;

    // workspace layout (floats); total 6,815,744 floats = ~26 MB
    float* w1m  = ws;                         // H*F = 1048576
    float* w2m  = w1m  + (size_t)HH * FF;     // C*H = 1048576
    float* mem1 = w2m  + (size_t)CC * HH;     // B*H = 2097152
    float* mem2 = mem1 + (size_t)BB * HH;     // B*C = 524288
    float* spk1 = mem2 + (size_t)BB * CC;     // B*H = 2097152

    // masked weights + zero membranes (deterministic: rebuilt every call)
    e2net_init<<<(BB * HH) / 256, 256, 0, stream>>>(w1, mask1, w2, mask2,
                                                    w1m, w2m, mem1, mem2);

    const int tiles1 = (BB / 32) * (HH / 32);   // 2048 wave-tiles
    const int tiles2 = (BB / 32) * (CC / 32);   // 512 wave-tiles
    const int wavesPerBlock = 8;                // 256 threads

    for (int t = 0; t < TT; ++t) {
        const float* xt = x + (size_t)t * BB * FF;
        // layer 1: cur = x_t @ w1m^T, update mem1, emit spk1 (launch = global sync)
        lif_gemm<<<(tiles1 + wavesPerBlock - 1) / wavesPerBlock, 256, 0, stream>>>(
            xt, w1m, mem1, spk1, BB, HH, FF);
        // layer 2: out = spk1 @ w2m^T, update mem2, emit spikes to d_out[t]
        lif_gemm<<<(tiles2 + wavesPerBlock - 1) / wavesPerBlock, 256, 0, stream>>>(
            spk1, w2m, mem2, out + (size_t)t * BB * CC, BB, CC, HH);
    }
}